// MaskPredictor_81011673137606
// MI455X (gfx1250) — compile-verified
//
#include <hip/hip_runtime.h>
#include <stdint.h>

#define B_   32
#define H_   12
#define BH_  (B_ * H_)          // 384
#define N_   577
#define CH_  64
#define RC_  32
#define RN_  72
#define M_   576                // N_ - 1 (rows after slicing off row 0)
#define MT_  36                 // M_ / 16
#define NT_  37                 // ceil(N_ / 16)
#define NPAD_ (NT_ * 16)        // 592 padded columns
#define KC_  (RN_ / 4)          // 18 K-chunks of 4
#define TOPK_    16
#define BUDGET_  145
#define SCALE_   0.28867513459481287f   // 12^-0.5
#define THR_     0.02f

typedef __attribute__((ext_vector_type(2))) float v2f;
typedef __attribute__((ext_vector_type(8))) float v8f;

// ---------------------------------------------------------------- packed basis
// WMMA-ready B layout. For K-chunk kc4 (4 K-values) and half h (0: K,K+1 for
// lanes 0-15 | 1: K+2,K+3 for lanes 16-31), store per column the float2
//   { basis[kc4*4 + 2h][col], basis[kc4*4 + 2h + 1][col] }
// where basis[r][col] = |proj_back[col][r]| thresholded, col >= 577 -> 0.
// pb index: ((kc4*2 + h) * NPAD_ + col) * 2
__global__ void mp_basis_pack_kernel(const float* __restrict__ proj_back,
                                     float* __restrict__ pb) {
    int i = blockIdx.x * blockDim.x + threadIdx.x;   // over KC_*2*NPAD_ pairs
    if (i >= KC_ * 2 * NPAD_) return;
    int g = i / NPAD_, col = i - g * NPAD_;
    int r0 = (g >> 1) * 4 + (g & 1) * 2;             // kc4*4 + h*2
    float f0 = 0.0f, f1 = 0.0f;
    if (col < N_) {
        float a0 = fabsf(proj_back[col * RN_ + r0]);
        float a1 = fabsf(proj_back[col * RN_ + r0 + 1]);
        f0 = (a0 > THR_) ? a0 : 0.0f;
        f1 = (a1 > THR_) ? a1 : 0.0f;
    }
    pb[2 * i]     = f0;
    pb[2 * i + 1] = f1;
}

// ---------------------------------------------------------------- CLS row
__global__ void mp_cls_kernel(float* __restrict__ attn) {
    long long bh = blockIdx.x;
    float* row = attn + bh * (long long)N_ * N_;     // row 0 of each head
    for (int c = threadIdx.x; c < N_; c += blockDim.x) row[c] = 1.0f;
}

// ---------------------------------------------------------------- qp / kp
// one 64-thread block per (bh, n) row; lanes 0..31 -> qp, 32..63 -> kp
__global__ void mp_qpkp_kernel(const float* __restrict__ q,
                               const float* __restrict__ k,
                               const float* __restrict__ Wq,
                               const float* __restrict__ bq,
                               const float* __restrict__ Wk,
                               const float* __restrict__ bk,
                               float* __restrict__ qp,
                               float* __restrict__ kp) {
    __shared__ float srow[2][CH_];
    long long row = blockIdx.x;                      // bh*N + n
    int t = threadIdx.x;
    srow[0][t] = q[row * CH_ + t];
    srow[1][t] = k[row * CH_ + t];
    __syncthreads();
    const float* W    = (t < RC_) ? Wq : Wk;
    const float* bias = (t < RC_) ? bq : bk;
    const float* x    = srow[(t < RC_) ? 0 : 1];
    int co = t & (RC_ - 1);
    float acc = bias[co];
#pragma unroll
    for (int c = 0; c < CH_; ++c) acc += x[c] * W[c * RC_ + co];
    float* dst = (t < RC_) ? qp : kp;
    dst[row * RC_ + co] = acc;
}

// ---------------------------------------------------------------- klow
// per head: klow[c][r] = sum_n kp[n][c] * proj_n[n][r]   (32 x 72)
__global__ void mp_klow_kernel(const float* __restrict__ kp,
                               const float* __restrict__ proj_n,
                               float* __restrict__ klow) {
    __shared__ float s_kp[64][RC_];   //  8 KB
    __shared__ float s_pn[64][RN_];   // 18 KB
    long long bh = blockIdx.x;
    const float* kpb = kp + bh * (long long)N_ * RC_;
    int t = threadIdx.x;
    float acc[9];
#pragma unroll
    for (int j = 0; j < 9; ++j) acc[j] = 0.0f;

    for (int n0 = 0; n0 < N_; n0 += 64) {
        int nn = N_ - n0; if (nn > 64) nn = 64;
        for (int e = t; e < 64 * RC_; e += 256) {
            int ln = e / RC_, c = e - ln * RC_;
            s_kp[ln][c] = (ln < nn) ? kpb[(long long)(n0 + ln) * RC_ + c] : 0.0f;
        }
        for (int e = t; e < 64 * RN_; e += 256) {
            int ln = e / RN_, r = e - ln * RN_;
            s_pn[ln][r] = (ln < nn) ? proj_n[(long long)(n0 + ln) * RN_ + r] : 0.0f;
        }
        __syncthreads();
#pragma unroll
        for (int j = 0; j < 9; ++j) {
            int o = t + 256 * j;
            int c = o / RN_, r = o - c * RN_;
            float a = acc[j];
            for (int ln = 0; ln < 64; ++ln) a += s_kp[ln][c] * s_pn[ln][r];
            acc[j] = a;
        }
        __syncthreads();
    }
#pragma unroll
    for (int j = 0; j < 9; ++j)
        klow[bh * (long long)(RC_ * RN_) + t + 256 * j] = acc[j];
}

// ---------------------------------------------------------------- coef
// per head; one wave per output row m (= token n = m+1):
//   cheap[r] = SCALE * dot(qp[n], klow[:,r]); softmax(72); keep top-16.
__global__ void mp_coef_kernel(const float* __restrict__ qp,
                               const float* __restrict__ klow,
                               float* __restrict__ coef) {
    __shared__ float s_kl[RC_ * RN_ + 64];
    long long bh = blockIdx.x;
    int t = threadIdx.x;
    for (int e = t; e < RC_ * RN_ + 64; e += 256)
        s_kl[e] = (e < RC_ * RN_) ? klow[bh * (long long)(RC_ * RN_) + e] : 0.0f;
    __syncthreads();

    int wave = t >> 5, lane = t & 31;
    int r0 = lane, r1 = lane + 32, r2 = lane + 64;
    bool v2ok = (lane < 8);                          // r2 < 72 only for lanes 0..7

    for (int m = wave; m < M_; m += 8) {
        int n = m + 1;
        float qv = qp[(bh * (long long)N_ + n) * RC_ + lane];
        float a0 = 0.f, a1 = 0.f, a2 = 0.f;
#pragma unroll
        for (int c = 0; c < RC_; ++c) {
            float qc = __shfl(qv, c, 32);
            const float* kl = &s_kl[c * RN_];
            a0 += qc * kl[r0];
            a1 += qc * kl[r1];
            a2 += qc * kl[r2];                       // pad region zeroed; masked later
        }
        a0 *= SCALE_; a1 *= SCALE_; a2 *= SCALE_;

        // ---- softmax over 72
        float mx = a0 > a1 ? a0 : a1;
        if (v2ok && a2 > mx) mx = a2;
        for (int off = 16; off; off >>= 1) {
            float o = __shfl_xor(mx, off, 32);
            if (o > mx) mx = o;
        }
        float e0 = expf(a0 - mx), e1 = expf(a1 - mx), e2 = expf(a2 - mx);
        float sm = e0 + e1 + (v2ok ? e2 : 0.0f);
        for (int off = 16; off; off >>= 1) sm += __shfl_xor(sm, off, 32);
        float inv = 1.0f / sm;
        float p0 = e0 * inv, p1 = e1 * inv, p2 = e2 * inv;

        // ---- top-16 (ties -> lowest index, matching jax top_k)
        float c0 = p0, c1 = p1, c2 = v2ok ? p2 : -1.0f;
        bool  s0 = false, s1 = false, s2 = false;
#pragma unroll 1
        for (int it = 0; it < TOPK_; ++it) {
            float bv = c0; int bi = r0;
            if (c1 > bv) { bv = c1; bi = r1; }
            if (c2 > bv) { bv = c2; bi = r2; }
            for (int off = 16; off; off >>= 1) {
                float ov = __shfl_xor(bv, off, 32);
                int   oi = __shfl_xor(bi, off, 32);
                if (ov > bv || (ov == bv && oi < bi)) { bv = ov; bi = oi; }
            }
            if      (bi == r0) { s0 = true; c0 = -1.0f; }
            else if (bi == r1) { s1 = true; c1 = -1.0f; }
            else if (bi == r2) { s2 = true; c2 = -1.0f; }
        }
        float* orow = coef + (bh * (long long)M_ + m) * RN_;
        orow[r0] = s0 ? p0 : 0.0f;
        orow[r1] = s1 ? p1 : 0.0f;
        if (v2ok) orow[r2] = s2 ? p2 : 0.0f;
    }
}

// ---------------------------------------------------------------- approx + mask
// grid (BH, 36). Block = 8 waves. Each wave: WMMA f32 16x16x4 tiles of
// approx = coef(576x72) @ basis(72x577) with branch-free b64 fragment loads
// from the packed basis; tile kept in LDS, then per-row radix-select of the
// 145th largest + index-ordered tie-fill -> attn_mask row.
__global__ void mp_approx_mask_kernel(const float* __restrict__ coef,
                                      const float* __restrict__ pb,
                                      float* __restrict__ approx,
                                      float* __restrict__ attn) {
    __shared__ float s_tile[16][NPAD_ + 8];          // 16 x 600 floats = 38.4 KB
    long long bh = blockIdx.x;
    int mt = blockIdx.y;
    int m0 = mt * 16;
    int wave = threadIdx.x >> 5, lane = threadIdx.x & 31;

    const float* Arow = coef + (bh * (long long)M_ + m0) * RN_;
    int am   = lane & 15;              // M (A) / N (B) position within tile
    int half = lane >> 4;              // 0: K,K+1   1: K+2,K+3

    // hoist A fragments (same 16x72 A tile for every N-tile of this block)
    v2f afrag[KC_];
#pragma unroll
    for (int kc4 = 0; kc4 < KC_; ++kc4)
        afrag[kc4] = *(const v2f*)(Arow + am * RN_ + kc4 * 4 + half * 2);

    float* gout = approx + bh * (long long)M_ * N_ + (long long)m0 * N_;
    int crow = half * 8;               // C/D: lanes16-31 -> M+8
    int ccol = lane & 15;

    for (int nt = wave; nt < NT_; nt += 8) {
        int n0 = nt * 16;
        int col = n0 + am;             // < NPAD_ always; pad cols are zero
        // branch-free B fragment loads (one b64 per lane per K-chunk)
        v2f bfrag[KC_];
#pragma unroll
        for (int kc4 = 0; kc4 < KC_; ++kc4)
            bfrag[kc4] = *(const v2f*)(pb + ((long long)(kc4 * 2 + half) * NPAD_ + col) * 2);

        v8f acc = {};
#pragma unroll
        for (int kc4 = 0; kc4 < KC_; ++kc4)
            acc = __builtin_amdgcn_wmma_f32_16x16x4_f32(
                      false, afrag[kc4], false, bfrag[kc4], (short)0, acc, false, false);

        int wcol = n0 + ccol;
#pragma unroll
        for (int j = 0; j < 8; ++j) {
            float v = acc[j];
            int rr = j + crow;
            s_tile[rr][wcol] = v;
            if (wcol < N_) gout[(long long)rr * N_ + wcol] = v;
        }
    }
    __syncthreads();

    // ---- per-row top-145 mask; wave handles 2 rows
    for (int lr = wave * 2; lr < wave * 2 + 2; ++lr) {
        int m = m0 + lr;               // 0..575 -> attn row m+1
        // radix select on bits (all values >= 0 -> bit order == value order)
        unsigned prefix = 0u;
        for (int bit = 30; bit >= 0; --bit) {
            unsigned cand = prefix | (1u << bit);
            int cnt = 0;
            for (int c = lane; c < N_; c += 32)
                cnt += (__float_as_uint(s_tile[lr][c]) >= cand) ? 1 : 0;
            for (int off = 16; off; off >>= 1) cnt += __shfl_xor(cnt, off, 32);
            if (cnt >= BUDGET_) prefix = cand;
        }
        float T = __uint_as_float(prefix);
        int cnt_gt = 0;
        for (int c = lane; c < N_; c += 32)
            cnt_gt += (s_tile[lr][c] > T) ? 1 : 0;
        for (int off = 16; off; off >>= 1) cnt_gt += __shfl_xor(cnt_gt, off, 32);
        int need_eq = BUDGET_ - cnt_gt;

        float* mrow = attn + (bh * (long long)N_ + (m + 1)) * N_;
        int run_eq = 0;
        for (int c0 = 0; c0 < N_; c0 += 32) {
            int c = c0 + lane;
            bool valid = (c < N_);
            float v = valid ? s_tile[lr][c] : -1.0f;
            bool eq = valid && (v == T);
            unsigned bal = (unsigned)__ballot(eq);
            int rank = run_eq + __popc(bal & ((1u << lane) - 1u));
            if (valid)
                mrow[c] = (v > T || (eq && rank < need_eq)) ? 1.0f : 0.0f;
            run_eq += __popc(bal);
        }
    }
}

// ---------------------------------------------------------------- launch
extern "C" void kernel_launch(void* const* d_in, const int* in_sizes, int n_in,
                              void* d_out, int out_size, void* d_ws, size_t ws_size,
                              hipStream_t stream) {
    const float* q         = (const float*)d_in[0];
    const float* k         = (const float*)d_in[1];
    const float* Wq        = (const float*)d_in[2];
    const float* bq        = (const float*)d_in[3];
    const float* Wk        = (const float*)d_in[4];
    const float* bk        = (const float*)d_in[5];
    const float* proj_n    = (const float*)d_in[6];
    const float* proj_back = (const float*)d_in[7];

    float* out    = (float*)d_out;
    float* coef   = out;                                   // 384*576*72
    float* approx = out + (long long)BH_ * M_ * RN_;       // +15,925,248
    float* attn   = approx + (long long)BH_ * M_ * N_;     // +127,623,168

    float* ws    = (float*)d_ws;
    float* qp    = ws;
    float* kp    = qp + (long long)BH_ * N_ * RC_;
    float* klow  = kp + (long long)BH_ * N_ * RC_;
    float* pb    = klow + (long long)BH_ * RC_ * RN_;      // KC_*2*NPAD_*2 floats

    mp_basis_pack_kernel<<<(KC_ * 2 * NPAD_ + 255) / 256, 256, 0, stream>>>(proj_back, pb);
    mp_cls_kernel<<<BH_, 256, 0, stream>>>(attn);
    mp_qpkp_kernel<<<BH_ * N_, 64, 0, stream>>>(q, k, Wq, bq, Wk, bk, qp, kp);
    mp_klow_kernel<<<BH_, 256, 0, stream>>>(kp, proj_n, klow);
    mp_coef_kernel<<<BH_, 256, 0, stream>>>(qp, klow, coef);
    mp_approx_mask_kernel<<<dim3(BH_, MT_), 256, 0, stream>>>(coef, pb, approx, attn);
}